// DihedralBiasVMap2_9113920602372
// MI455X (gfx1250) — compile-verified
//
#include <hip/hip_runtime.h>
#include <math.h>

// ---------------- problem constants ----------------
#define M_MODELS 32
#define N_CVS    4000
#define FEAT     512
#define IN_DIM   8000          // 2*N_CVS
#define NPART    1000000
#define OUT_N    3000001       // 1 scalar energy + 1M*3 forces
#define L1_CHUNKS 16
#define L1_CHUNK_I 500
#define PI_F 3.14159265358979f

// ---------------- workspace layout (floats) ----------------
#define WS_XCS    0                         // 8000: cos||sin
#define WS_PART   (WS_XCS + IN_DIM)         // 32*512*16 split-K partials
#define WS_H1     (WS_PART + M_MODELS*FEAT*L1_CHUNKS)
#define WS_H2     (WS_H1 + M_MODELS*FEAT)
#define WS_H3     (WS_H2 + M_MODELS*FEAT)
#define WS_H4     (WS_H3 + M_MODELS*FEAT)
#define WS_GZ1    (WS_H4 + M_MODELS*FEAT)   // final layer-1 input gradient
#define WS_GZA    (WS_GZ1 + M_MODELS*FEAT)  // gradient ping
#define WS_GZB    (WS_GZA + M_MODELS*FEAT)  // gradient pong
#define WS_GX     (WS_GZB + M_MODELS*FEAT)  // 32*8000
#define WS_E      (WS_GX + M_MODELS*IN_DIM) // 32 energies
#define WS_VARC   (WS_E + M_MODELS)         // 4000 per-cv variances
#define WS_GMT    (WS_VARC + N_CVS)         // 4000 mean d(meanE)/dtors
#define WS_SIGMA  (WS_GMT + N_CVS)          // 1

typedef float v2f __attribute__((ext_vector_type(2)));
typedef float v8f __attribute__((ext_vector_type(8)));

struct f3 { float x, y, z; };
__device__ __forceinline__ f3 f3sub(f3 a, f3 b){ return {a.x-b.x, a.y-b.y, a.z-b.z}; }
__device__ __forceinline__ f3 f3cross(f3 a, f3 b){
    return {a.y*b.z-a.z*b.y, a.z*b.x-a.x*b.z, a.x*b.y-a.y*b.x};
}
__device__ __forceinline__ float f3dot(f3 a, f3 b){ return a.x*b.x+a.y*b.y+a.z*b.z; }
__device__ __forceinline__ f3 f3scale(f3 a, float s){ return {a.x*s, a.y*s, a.z*s}; }

__device__ __forceinline__ f3 load_wrapped(const float* pos, int id, float bx, float by, float bz){
    float px = pos[(size_t)id*3+0], py = pos[(size_t)id*3+1], pz = pos[(size_t)id*3+2];
    px -= floorf(px/bx)*bx; py -= floorf(py/by)*by; pz -= floorf(pz/bz)*bz;
    return {px, py, pz};
}

// ---------------- 0: zero the output (forces accumulate atomically) ----------------
__global__ void k_zero(float* out){
    long i = (long)blockIdx.x*blockDim.x + threadIdx.x;
    long stride = (long)gridDim.x*blockDim.x;
    for (long j = i; j < OUT_N; j += stride) out[j] = 0.f;
}

// ---------------- 1: torsions -> cos/sin feature vector ----------------
__global__ void k_tors(const float* __restrict__ pos, const float* __restrict__ boxv,
                       const int* __restrict__ cidx, float* __restrict__ xcs){
    int c = blockIdx.x*blockDim.x + threadIdx.x;
    if (c >= N_CVS) return;
    float bx = boxv[0], by = boxv[4], bz = boxv[8];
    f3 p0 = load_wrapped(pos, cidx[c*4+0], bx, by, bz);
    f3 p1 = load_wrapped(pos, cidx[c*4+1], bx, by, bz);
    f3 p2 = load_wrapped(pos, cidx[c*4+2], bx, by, bz);
    f3 p3 = load_wrapped(pos, cidx[c*4+3], bx, by, bz);
    f3 b1 = f3sub(p1, p0), b2 = f3sub(p2, p1), b3 = f3sub(p3, p2);
    f3 n1 = f3cross(b1, b2), n2 = f3cross(b2, b3);
    float ib2 = rsqrtf(f3dot(b2, b2));
    f3 b2u = f3scale(b2, ib2);
    f3 m1 = f3cross(n1, b2u);
    float yy = f3dot(m1, n2);          // sin * r
    float xx = f3dot(n1, n2);          // cos * r
    float ir = rsqrtf(xx*xx + yy*yy);
    xcs[c]         = xx * ir;          // cos(t)
    xcs[N_CVS + c] = yy * ir;          // sin(t)
}

// ---------------- 2: layer-1 forward, deterministic split-K ----------------
// grid: 512 blocks = 32 models x 16 i-chunks, blockDim 512 (one thread per output o)
__global__ __launch_bounds__(512) void k_l1_fwd(const float* __restrict__ W1,
                                                const float* __restrict__ xcs,
                                                float* __restrict__ partial){
    int m  = blockIdx.x >> 4;
    int ch = blockIdx.x & 15;
    int o  = threadIdx.x;
    const float* Wp = W1 + ((size_t)m*IN_DIM + (size_t)ch*L1_CHUNK_I)*FEAT + o;
    const float* xp = xcs + ch*L1_CHUNK_I;
    float acc = 0.f;
    #pragma unroll 4
    for (int i = 0; i < L1_CHUNK_I; ++i)
        acc = fmaf(xp[i], Wp[(size_t)i*FEAT], acc);
    partial[(((size_t)m*FEAT + o) << 4) + ch] = acc;
}

// ---------------- 3: reduce split-K (fixed order) + bias + relu ----------------
__global__ void k_relu1(const float* __restrict__ partial, const float* __restrict__ b1,
                        float* __restrict__ h1){
    int idx = blockIdx.x*blockDim.x + threadIdx.x;
    if (idx >= M_MODELS*FEAT) return;
    float s = 0.f;
    #pragma unroll
    for (int ch = 0; ch < L1_CHUNKS; ++ch) s += partial[((size_t)idx << 4) + ch];
    s += b1[idx];
    h1[idx] = s > 0.f ? s : 0.f;
}

// ---------------- 4: one 512x512 layer forward via f32 WMMA (broadcast-A GEMV) ----------------
// grid: 256 blocks = 32 models x 8 o-blocks of 64 outputs; blockDim 128 (4 waves).
// Wave w computes output tile o0 = ob*64 + 16*w.  D = A(bcast h rows) x B(W tile).
__global__ void __launch_bounds__(128) k_mid_fwd(
        const float* __restrict__ hin, const float* __restrict__ W,
        const float* __restrict__ b, float* __restrict__ hout){
    __shared__ float hs[FEAT];
    int m = blockIdx.x >> 3, ob = blockIdx.x & 7;
    int tid = threadIdx.x, wave = tid >> 5, lane = tid & 31;
    int n = lane & 15, kh = lane >> 4;          // kh: which K-pair this half-wave holds
    for (int t = tid; t < FEAT; t += 128) hs[t] = hin[(size_t)m*FEAT + t];
    __syncthreads();

    const float* Wm = W + (size_t)m*FEAT*FEAT;
    int o0 = ob*64 + (wave << 4);
    v8f acc = {};
    for (int k0 = 0; k0 < FEAT; k0 += 4){
        v2f a, bb;
        a.x = hs[k0 + 2*kh];                    // A: 16 identical rows (matvec broadcast)
        a.y = hs[k0 + 2*kh + 1];
        const float* Bp = Wm + (size_t)(k0 + 2*kh)*FEAT + o0 + n;
        bb.x = Bp[0];                           // B: W[k, o0+n]
        bb.y = Bp[FEAT];
        acc = __builtin_amdgcn_wmma_f32_16x16x4_f32(false, a, false, bb,
                                                    (short)0, acc, false, false);
    }
    if (lane < 16){                             // row M=0 of D (all rows equal)
        float v = acc[0] + b[(size_t)m*FEAT + o0 + lane];
        hout[(size_t)m*FEAT + o0 + lane] = v > 0.f ? v : 0.f;
    }
}

// ---------------- 5: layer-5 energy: e_m = dot(h4, W5[m]) + b5[m] ----------------
__global__ __launch_bounds__(256) void k_e5(const float* __restrict__ h4,
                                            const float* __restrict__ W5,
                                            const float* __restrict__ b5,
                                            float* __restrict__ eg){
    __shared__ float red[256];
    int m = blockIdx.x, tid = threadIdx.x;
    const float* h = h4 + (size_t)m*FEAT;
    const float* w = W5 + (size_t)m*FEAT;
    red[tid] = h[tid]*w[tid] + h[tid+256]*w[tid+256];   // fixed pairing: deterministic
    __syncthreads();
    for (int s = 128; s > 0; s >>= 1){
        if (tid < s) red[tid] += red[tid + s];
        __syncthreads();
    }
    if (tid == 0) eg[m] = red[0] + b5[m];
}

// ---------------- 6: seed backward: g_z4 = W5 * relu'(h4) ----------------
__global__ void k_gz4(const float* __restrict__ W5, const float* __restrict__ h4,
                      float* __restrict__ g4){
    int idx = blockIdx.x*blockDim.x + threadIdx.x;
    if (idx >= M_MODELS*FEAT) return;
    g4[idx] = W5[idx] * (h4[idx] > 0.f ? 1.f : 0.f);
}

// ---------------- 7: one 512x512 layer backward via f32 WMMA (W-tile A, broadcast-B) ----------
// gout[i] = relu'(hprev[i]) * sum_o W[i,o] * gin[o].
// grid: 256 blocks = 32 models x 8 i-blocks of 64 rows; blockDim 128 (4 waves).
__global__ void __launch_bounds__(128) k_mid_bwd(
        const float* __restrict__ gin, const float* __restrict__ W,
        const float* __restrict__ hprev, float* __restrict__ gout){
    __shared__ float gs[FEAT];
    int m = blockIdx.x >> 3, ib = blockIdx.x & 7;
    int tid = threadIdx.x, wave = tid >> 5, lane = tid & 31;
    int r = lane & 15, kh = lane >> 4;
    for (int t = tid; t < FEAT; t += 128) gs[t] = gin[(size_t)m*FEAT + t];
    __syncthreads();

    const float* Wm = W + (size_t)m*FEAT*FEAT;
    int i0 = ib*64 + (wave << 4);               // 16 input-gradient rows per wave
    v8f acc = {};
    for (int k0 = 0; k0 < FEAT; k0 += 4){
        v2f a, bb;
        const float* Ap = Wm + (size_t)(i0 + r)*FEAT + k0 + 2*kh;   // A = W[i, o]
        a.x = Ap[0];
        a.y = Ap[1];
        bb.x = gs[k0 + 2*kh];                   // B: broadcast gradient columns
        bb.y = gs[k0 + 2*kh + 1];
        acc = __builtin_amdgcn_wmma_f32_16x16x4_f32(false, a, false, bb,
                                                    (short)0, acc, false, false);
    }
    if (r == 0){                                // lanes 0 (M=0..7) and 16 (M=8..15)
        int ibase = i0 + (kh << 3);
        #pragma unroll
        for (int v = 0; v < 8; ++v){
            int i = ibase + v;
            float val = acc[v] * (hprev[(size_t)m*FEAT + i] > 0.f ? 1.f : 0.f);
            gout[(size_t)m*FEAT + i] = val;
        }
    }
}

// ---------------- 8: layer-1 backward: gx[m,i] = sum_o W1[m,i,o]*gz1[m,o] ----------------
// One wave per (m,i): coalesced 2KB row stream + shfl reduce. 256000 waves.
__global__ __launch_bounds__(256) void k_l1_bwd(const float* __restrict__ W1,
                                                const float* __restrict__ gz1,
                                                float* __restrict__ gx){
    int wid  = blockIdx.x*(blockDim.x >> 5) + (threadIdx.x >> 5);
    int lane = threadIdx.x & 31;
    int m = wid / IN_DIM, i = wid - m*IN_DIM;
    const float* Wp = W1 + ((size_t)m*IN_DIM + i)*FEAT;
    const float* gp = gz1 + (size_t)m*FEAT;
    float acc = 0.f;
    #pragma unroll
    for (int j = 0; j < FEAT/32; ++j)
        acc = fmaf(Wp[lane + 32*j], gp[lane + 32*j], acc);
    for (int off = 16; off > 0; off >>= 1) acc += __shfl_xor(acc, off, 32);
    if (lane == 0) gx[(size_t)m*IN_DIM + i] = acc;
}

// ---------------- 9: per-CV jacobian stats: mean over models + unbiased variance ----------------
__global__ void k_jac(const float* __restrict__ gx, const float* __restrict__ xcs,
                      float* __restrict__ varc, float* __restrict__ gmt){
    int c = blockIdx.x*blockDim.x + threadIdx.x;
    if (c >= N_CVS) return;
    float cs = xcs[c], sn = xcs[N_CVS + c];
    float s = 0.f, s2 = 0.f;
    #pragma unroll 4
    for (int m = 0; m < M_MODELS; ++m){
        float j = -sn*gx[(size_t)m*IN_DIM + c] + cs*gx[(size_t)m*IN_DIM + N_CVS + c];
        s += j; s2 += j*j;
    }
    float mean = s * (1.f/M_MODELS);
    varc[c] = (s2 - (float)M_MODELS*mean*mean) * (1.f/(M_MODELS-1));
    gmt[c]  = mean;
}

// ---------------- 10: finalize scalars (deterministic tree reduce) ----------------
__global__ __launch_bounds__(256) void k_final(const float* __restrict__ varc,
                                               const float* __restrict__ eg,
                                               float* __restrict__ sigp,
                                               float* __restrict__ out){
    __shared__ float red[256];
    int tid = threadIdx.x;
    float s = 0.f;
    for (int c = tid; c < N_CVS; c += 256) s += varc[c];   // fixed per-thread order
    red[tid] = s;
    __syncthreads();
    for (int st = 128; st > 0; st >>= 1){
        if (tid < st) red[tid] += red[tid + st];
        __syncthreads();
    }
    if (tid == 0){
        float model_div = sqrtf(red[0] * (1.f/N_CVS));
        float esum = 0.f;
        for (int m = 0; m < M_MODELS; ++m) esum += eg[m];
        float energy_ave = esum * (1.f/M_MODELS);
        const float E0 = 2.f, E1 = 3.f;
        float isw = (E1 - model_div) / (E1 - E0);
        float smooth = 0.5f * (1.f + cosf(PI_F * (1.f - isw)));
        float fl = floorf(isw);
        float sigma = (fl > 0.f) ? 1.f : ((fl < 0.f) ? 0.f : smooth);
        sigp[0] = sigma;
        out[0]  = energy_ave * sigma;
    }
}

// ---------------- 11: analytic dihedral force scatter (Blondel–Karplus) ----------------
__global__ void k_forces(const float* __restrict__ pos, const float* __restrict__ boxv,
                         const int* __restrict__ cidx, const float* __restrict__ gmt,
                         const float* __restrict__ sigp, float* __restrict__ out){
    int c = blockIdx.x*blockDim.x + threadIdx.x;
    if (c >= N_CVS) return;
    float sig = sigp[0];
    float bx = boxv[0], by = boxv[4], bz = boxv[8];
    int i0 = cidx[c*4+0], i1 = cidx[c*4+1], i2 = cidx[c*4+2], i3 = cidx[c*4+3];
    f3 p0 = load_wrapped(pos, i0, bx, by, bz);
    f3 p1 = load_wrapped(pos, i1, bx, by, bz);
    f3 p2 = load_wrapped(pos, i2, bx, by, bz);
    f3 p3 = load_wrapped(pos, i3, bx, by, bz);
    f3 b1 = f3sub(p1, p0), b2 = f3sub(p2, p1), b3 = f3sub(p3, p2);
    f3 n1 = f3cross(b1, b2), n2 = f3cross(b2, b3);
    float n1sq = f3dot(n1, n1), n2sq = f3dot(n2, n2), b2sq = f3dot(b2, b2);
    float nb2 = sqrtf(b2sq);
    float cf = gmt[c] * sig;                    // dE_mean/dphi * sigma
    f3 f = f3scale(n1, -nb2/n1sq * cf);         // cf * dphi/dp0
    f3 g = f3scale(n2,  nb2/n2sq * cf);         // cf * dphi/dp3
    float t1 = f3dot(b1, b2)/b2sq, t2 = f3dot(b3, b2)/b2sq;
    f3 d0 = f;
    f3 d1 = {(t1-1.f)*f.x - t2*g.x, (t1-1.f)*f.y - t2*g.y, (t1-1.f)*f.z - t2*g.z};
    f3 d2 = {(t2-1.f)*g.x - t1*f.x, (t2-1.f)*g.y - t1*f.y, (t2-1.f)*g.z - t1*f.z};
    f3 d3 = g;
    float* F = out + 1;
    atomicAdd(&F[(size_t)i0*3+0], d0.x); atomicAdd(&F[(size_t)i0*3+1], d0.y); atomicAdd(&F[(size_t)i0*3+2], d0.z);
    atomicAdd(&F[(size_t)i1*3+0], d1.x); atomicAdd(&F[(size_t)i1*3+1], d1.y); atomicAdd(&F[(size_t)i1*3+2], d1.z);
    atomicAdd(&F[(size_t)i2*3+0], d2.x); atomicAdd(&F[(size_t)i2*3+1], d2.y); atomicAdd(&F[(size_t)i2*3+2], d2.z);
    atomicAdd(&F[(size_t)i3*3+0], d3.x); atomicAdd(&F[(size_t)i3*3+1], d3.y); atomicAdd(&F[(size_t)i3*3+2], d3.z);
}

// ---------------- host launch ----------------
extern "C" void kernel_launch(void* const* d_in, const int* in_sizes, int n_in,
                              void* d_out, int out_size, void* d_ws, size_t ws_size,
                              hipStream_t stream){
    const float* pos  = (const float*)d_in[0];
    const float* boxv = (const float*)d_in[1];
    const int*   cidx = (const int*)  d_in[2];
    const float* W1 = (const float*)d_in[3];  const float* b1 = (const float*)d_in[4];
    const float* W2 = (const float*)d_in[5];  const float* b2 = (const float*)d_in[6];
    const float* W3 = (const float*)d_in[7];  const float* b3 = (const float*)d_in[8];
    const float* W4 = (const float*)d_in[9];  const float* b4 = (const float*)d_in[10];
    const float* W5 = (const float*)d_in[11]; const float* b5 = (const float*)d_in[12];
    float* out = (float*)d_out;
    float* ws  = (float*)d_ws;

    float* xcs  = ws + WS_XCS;
    float* part = ws + WS_PART;
    float* h1   = ws + WS_H1;
    float* h2   = ws + WS_H2;
    float* h3   = ws + WS_H3;
    float* h4   = ws + WS_H4;
    float* gz1  = ws + WS_GZ1;
    float* gza  = ws + WS_GZA;
    float* gzb  = ws + WS_GZB;
    float* gx   = ws + WS_GX;
    float* eg   = ws + WS_E;
    float* varc = ws + WS_VARC;
    float* gmt  = ws + WS_GMT;
    float* sigp = ws + WS_SIGMA;

    k_zero<<<2048, 256, 0, stream>>>(out);
    k_tors<<<(N_CVS + 255)/256, 256, 0, stream>>>(pos, boxv, cidx, xcs);

    // forward
    k_l1_fwd<<<M_MODELS*L1_CHUNKS, 512, 0, stream>>>(W1, xcs, part);
    k_relu1<<<(M_MODELS*FEAT + 255)/256, 256, 0, stream>>>(part, b1, h1);
    k_mid_fwd<<<M_MODELS*8, 128, 0, stream>>>(h1, W2, b2, h2);
    k_mid_fwd<<<M_MODELS*8, 128, 0, stream>>>(h2, W3, b3, h3);
    k_mid_fwd<<<M_MODELS*8, 128, 0, stream>>>(h3, W4, b4, h4);
    k_e5<<<M_MODELS, 256, 0, stream>>>(h4, W5, b5, eg);

    // backward (per-model jacobians; also yields mean-energy gradient)
    k_gz4<<<(M_MODELS*FEAT + 255)/256, 256, 0, stream>>>(W5, h4, gza);
    k_mid_bwd<<<M_MODELS*8, 128, 0, stream>>>(gza, W4, h3, gzb);
    k_mid_bwd<<<M_MODELS*8, 128, 0, stream>>>(gzb, W3, h2, gza);
    k_mid_bwd<<<M_MODELS*8, 128, 0, stream>>>(gza, W2, h1, gz1);
    k_l1_bwd<<<(M_MODELS*IN_DIM)/8, 256, 0, stream>>>(W1, gz1, gx);

    // statistics, switch, outputs
    k_jac<<<(N_CVS + 255)/256, 256, 0, stream>>>(gx, xcs, varc, gmt);
    k_final<<<1, 256, 0, stream>>>(varc, eg, sigp, out);
    k_forces<<<(N_CVS + 255)/256, 256, 0, stream>>>(pos, boxv, cidx, gmt, sigp, out);
}